// Mutag_GCN2_26371099198072
// MI455X (gfx1250) — compile-verified
//
#include <hip/hip_runtime.h>
#include <math.h>

typedef __attribute__((ext_vector_type(2))) float v2f;
typedef __attribute__((ext_vector_type(8))) float v8f;

#define HID    30
#define HP     32          // padded hidden stride
#define NGRAPH 2048
#define POOLW  120         // 4 * HID

// ---------------------------------------------------------------- utilities
__device__ __forceinline__ void atomicMaxF(float* addr, float val) {
  // Order-independent float max via monotone int/uint mappings.
  if (val >= 0.0f) atomicMax((int*)addr, __float_as_int(val));
  else             atomicMin((unsigned int*)addr, __float_as_uint(val));
}

// ------------------------------------------------------------- prep kernels
__global__ void k_pad_weight(const float* __restrict__ W, float* __restrict__ Wp,
                             int srcRows, int outRows) {
  int idx = blockIdx.x * blockDim.x + threadIdx.x;
  if (idx >= outRows * HP) return;
  int r = idx >> 5, c = idx & 31;
  Wp[idx] = (r < srcRows && c < HID) ? W[r * HID + c] : 0.0f;
}

__global__ void k_init_deg(float* __restrict__ deg, int N) {
  int i = blockIdx.x * blockDim.x + threadIdx.x;
  if (i < N) deg[i] = 1.0f;                       // self-loop
}

__global__ void k_count_deg(const long long* __restrict__ ei, float* __restrict__ deg,
                            int E) {
  int e = blockIdx.x * blockDim.x + threadIdx.x;
  if (e < E) atomicAdd(&deg[(int)ei[(size_t)E + e]], 1.0f);   // col = dst
}

__global__ void k_finish_dis(float* __restrict__ dis, int N) {
  int i = blockIdx.x * blockDim.x + threadIdx.x;
  if (i >= N) return;
  float d = dis[i];
  dis[i] = (d > 0.0f) ? 1.0f / sqrtf(d) : 0.0f;
}

__global__ void k_init_pool(float* __restrict__ pool, int n) {
  int i = blockIdx.x * blockDim.x + threadIdx.x;
  if (i < n) pool[i] = -__builtin_inff();
}

// ------------------------------------------------------------ WMMA fp32 GEMM
// C[N x 32] = A[N x lda(K)] * B[K x 32], B pre-padded. One wave -> 16x32 tile.
// A row index is CLAMPED (not predicated) so the inner loop is branch-free;
// out-of-range rows compute garbage that the guarded store discards.
__global__ __launch_bounds__(256)
void k_gemm_wmma(const float* __restrict__ A, int lda, int K,
                 const float* __restrict__ B, float* __restrict__ C, int N) {
  int wave = blockIdx.x * (blockDim.x >> 5) + (threadIdx.x >> 5);
  int lane = threadIdx.x & 31;
  int m    = lane & 15;           // A row within tile (both lane halves)
  int kb   = (lane >> 4) << 1;    // K sub-offset: 0 (lanes 0-15) or 2 (16-31)
  int row  = wave * 16 + m;
  int rowc = row < N ? row : (N - 1);          // clamp: always a valid address
  const float* arow = A + (size_t)rowc * lda;

  v8f acc0 = {0.f,0.f,0.f,0.f,0.f,0.f,0.f,0.f};
  v8f acc1 = {0.f,0.f,0.f,0.f,0.f,0.f,0.f,0.f};

  for (int k0 = 0; k0 < K; k0 += 4) {
    float2 t = *reinterpret_cast<const float2*>(arow + k0 + kb); // 8B aligned
    v2f av; av.x = t.x; av.y = t.y;
    int kk = k0 + kb;
    v2f bv0, bv1;                 // B 4x16 frag: V0 holds K=kk, V1 holds K=kk+1
    bv0.x = B[kk * HP + m];       bv0.y = B[(kk + 1) * HP + m];
    bv1.x = B[kk * HP + m + 16];  bv1.y = B[(kk + 1) * HP + m + 16];
    acc0 = __builtin_amdgcn_wmma_f32_16x16x4_f32(false, av, false, bv0,
                                                 (short)0, acc0, false, false);
    acc1 = __builtin_amdgcn_wmma_f32_16x16x4_f32(false, av, false, bv1,
                                                 (short)0, acc1, false, false);
  }

  // D layout: VGPR r -> M = r + 8*(lane>=16), Ncol = lane&15
  int half = lane >> 4;
  int col  = lane & 15;
  for (int r = 0; r < 8; ++r) {
    int rr = wave * 16 + r + half * 8;
    if (rr < N) {
      C[(size_t)rr * HP + col]      = acc0[r];
      C[(size_t)rr * HP + col + 16] = acc1[r];
    }
  }
}

// --------------------------------------------------------- message passing
// h_out[i,h] = xw[i,h]*dis[i]^2 + b[h]   (self-loop folded into init)
__global__ void k_init_out(const float* __restrict__ xw, const float* __restrict__ dis,
                           const float* __restrict__ b, float* __restrict__ hout,
                           int N) {
  int idx = blockIdx.x * blockDim.x + threadIdx.x;
  if (idx >= N * HP) return;
  int i = idx >> 5, h = idx & 31;
  float v = 0.0f;
  if (h < HID) {
    float ds = dis[i];
    v = xw[idx] * ds * ds + b[h];
  }
  hout[idx] = v;
}

// one wave per edge, 30 active lanes: hout[col,:] += xw[row,:] * dis[row]*dis[col]
__global__ __launch_bounds__(256)
void k_scatter(const long long* __restrict__ ei, const float* __restrict__ dis,
               const float* __restrict__ xw, float* __restrict__ hout, int E) {
  long long gid = (long long)blockIdx.x * blockDim.x + threadIdx.x;
  int e = (int)(gid >> 5);
  int lane = (int)(gid & 31);
  if (e >= E) return;
  int r = (int)ei[e];
  int c = (int)ei[(size_t)E + e];
  float nrm = dis[r] * dis[c];
  if (lane < HID) {
    float v = xw[(size_t)r * HP + lane] * nrm;
    atomicAdd(&hout[(size_t)c * HP + lane], v);
  }
}

// fused (optional relu writeback) + segment-max pooling
__global__ __launch_bounds__(256)
void k_pool_relu(float* __restrict__ h, const long long* __restrict__ batch,
                 float* __restrict__ pool, int base, int doRelu, int N) {
  long long gid = (long long)blockIdx.x * blockDim.x + threadIdx.x;
  int i = (int)(gid >> 5);
  int lane = (int)(gid & 31);
  if (i >= N || lane >= HID) return;
  float v = h[(size_t)i * HP + lane];
  if (doRelu) {
    v = fmaxf(v, 0.0f);
    h[(size_t)i * HP + lane] = v;
  }
  int g = (int)batch[i];
  atomicMaxF(&pool[(size_t)g * POOLW + base + lane], v);
}

// ------------------------------------------------------------- final linear
__global__ void k_final(const float* __restrict__ pool, const float* __restrict__ Wlin,
                        const float* __restrict__ blin, float* __restrict__ out,
                        int total) {
  int idx = blockIdx.x * blockDim.x + threadIdx.x;
  if (idx >= total) return;
  int g = idx >> 1, c = idx & 1;
  float s = blin[c];
  const float* pg = pool + (size_t)g * POOLW;
  for (int j = 0; j < POOLW; ++j) {
    float v = pg[j];
    if (v == -__builtin_inff()) v = 0.0f;   // empty-segment fixup (reference)
    s += v * Wlin[j * 2 + c];
  }
  out[idx] = s;
}

// ------------------------------------------------------------------- driver
extern "C" void kernel_launch(void* const* d_in, const int* in_sizes, int n_in,
                              void* d_out, int out_size, void* d_ws, size_t ws_size,
                              hipStream_t stream) {
  const float*     x     = (const float*)d_in[0];
  const long long* ei    = (const long long*)d_in[1];
  const long long* batch = (const long long*)d_in[2];
  const float* W0 = (const float*)d_in[4];  const float* b0 = (const float*)d_in[5];
  const float* W1 = (const float*)d_in[6];  const float* b1 = (const float*)d_in[7];
  const float* W2 = (const float*)d_in[8];  const float* b2 = (const float*)d_in[9];
  const float* W3 = (const float*)d_in[10]; const float* b3 = (const float*)d_in[11];
  const float* Wlin = (const float*)d_in[12];
  const float* blin = (const float*)d_in[13];
  float* out = (float*)d_out;

  const int F = 128;
  int N = in_sizes[0] / F;           // 100000
  int E = in_sizes[1] / 2;           // 1600000
  size_t N32 = (size_t)N * HP;

  // workspace layout (floats)
  float* ws   = (float*)d_ws;
  float* dis  = ws;                                 // N
  float* xw   = dis + (((size_t)N + 63) / 64) * 64; // N*32
  float* hA   = xw + N32;                           // N*32
  float* hB   = hA + N32;                           // N*32
  float* pool = hB + N32;                           // NGRAPH*POOLW
  float* W0p  = pool + (size_t)NGRAPH * POOLW;      // 128*32
  float* W1p  = W0p + 128 * HP;                     // 32*32
  float* W2p  = W1p + 32 * HP;
  float* W3p  = W2p + 32 * HP;

  auto cdiv = [](long long a, long long b) { return (int)((a + b - 1) / b); };
  const int BT = 256;

  // weight padding
  k_pad_weight<<<cdiv(128 * HP, BT), BT, 0, stream>>>(W0, W0p, 128, 128);
  k_pad_weight<<<cdiv(32 * HP, BT),  BT, 0, stream>>>(W1, W1p, HID, 32);
  k_pad_weight<<<cdiv(32 * HP, BT),  BT, 0, stream>>>(W2, W2p, HID, 32);
  k_pad_weight<<<cdiv(32 * HP, BT),  BT, 0, stream>>>(W3, W3p, HID, 32);

  // gcn norm
  k_init_deg  <<<cdiv(N, BT), BT, 0, stream>>>(dis, N);
  k_count_deg <<<cdiv(E, BT), BT, 0, stream>>>(ei, dis, E);
  k_finish_dis<<<cdiv(N, BT), BT, 0, stream>>>(dis, N);
  k_init_pool <<<cdiv((long long)NGRAPH * POOLW, BT), BT, 0, stream>>>(
      pool, NGRAPH * POOLW);

  int gemmBlocks = cdiv(cdiv(N, 16), BT / 32);
  int nodeWaveBlocks = cdiv((long long)N * 32, BT);
  int edgeWaveBlocks = cdiv((long long)E * 32, BT);

  // layer 0: x[N,128] -> hA
  k_gemm_wmma<<<gemmBlocks, BT, 0, stream>>>(x, F, F, W0p, xw, N);
  k_init_out <<<nodeWaveBlocks, BT, 0, stream>>>(xw, dis, b0, hA, N);
  k_scatter  <<<edgeWaveBlocks, BT, 0, stream>>>(ei, dis, xw, hA, E);
  k_pool_relu<<<nodeWaveBlocks, BT, 0, stream>>>(hA, batch, pool, 0, 1, N);

  // layer 1: hA -> hB
  k_gemm_wmma<<<gemmBlocks, BT, 0, stream>>>(hA, HP, HP, W1p, xw, N);
  k_init_out <<<nodeWaveBlocks, BT, 0, stream>>>(xw, dis, b1, hB, N);
  k_scatter  <<<edgeWaveBlocks, BT, 0, stream>>>(ei, dis, xw, hB, E);
  k_pool_relu<<<nodeWaveBlocks, BT, 0, stream>>>(hB, batch, pool, 30, 1, N);

  // layer 2: hB -> hA
  k_gemm_wmma<<<gemmBlocks, BT, 0, stream>>>(hB, HP, HP, W2p, xw, N);
  k_init_out <<<nodeWaveBlocks, BT, 0, stream>>>(xw, dis, b2, hA, N);
  k_scatter  <<<edgeWaveBlocks, BT, 0, stream>>>(ei, dis, xw, hA, E);
  k_pool_relu<<<nodeWaveBlocks, BT, 0, stream>>>(hA, batch, pool, 60, 1, N);

  // layer 3 (no relu): hA -> hB
  k_gemm_wmma<<<gemmBlocks, BT, 0, stream>>>(hA, HP, HP, W3p, xw, N);
  k_init_out <<<nodeWaveBlocks, BT, 0, stream>>>(xw, dis, b3, hB, N);
  k_scatter  <<<edgeWaveBlocks, BT, 0, stream>>>(ei, dis, xw, hB, E);
  k_pool_relu<<<nodeWaveBlocks, BT, 0, stream>>>(hB, batch, pool, 90, 0, N);

  // readout @ Wlin + blin
  k_final<<<cdiv(NGRAPH * 2, BT), BT, 0, stream>>>(pool, Wlin, blin, out,
                                                   NGRAPH * 2);
}